// MultiScaleEdgeBuilder_67095979099194
// MI455X (gfx1250) — compile-verified
//
#include <hip/hip_runtime.h>
#include <hip/hip_bf16.h>

// MultiScaleEdgeBuilder: pairwise mask + RBF features.
//   out[0 .. N*N)            : mask as float (1.0 / 0.0), row-major [i][j]
//   out[N*N .. N*N + N*N*16) : rbf float32, [i][j][r]
//
// dist^2 via Gram matrix on the matrix unit:
//   d2(i,j) = |p_i|^2 + |p_j|^2 - 2 * (p_i . p_j)
// p_i . p_j for a 16x16 tile = one V_WMMA_F32_16X16X4_F32 (K=3 padded to 4).
//
// Output is 285 MB/call, write-only, never re-read -> non-temporal stores
// (TH=NT) so the stream doesn't thrash the 192 MB L2. ALU kept minimal
// (raw v_sqrt_f32 / v_rcp_f32, one v_exp_f32 per RBF) so the kernel stays
// on the 23.3 TB/s store roofline (~12 us floor).

#define NN 2048
#define NRBF 16
#define TILES_PER_ROW (NN / 16)        // 128
#define WAVES_PER_BLOCK 8

typedef __attribute__((ext_vector_type(2))) float v2f;
typedef __attribute__((ext_vector_type(4))) float v4f;   // native vector for NT stores
typedef __attribute__((ext_vector_type(8))) float v8f;

__global__ __launch_bounds__(256) void edge_rbf_wmma_kernel(
    const float* __restrict__ pos,      // [N,3]
    const int*   __restrict__ batch,    // [N]
    const float* __restrict__ centers,  // [16]
    const float* __restrict__ width,    // [1]
    float* __restrict__ outMask,        // [N,N]
    float* __restrict__ outRbf)         // [N,N,16]
{
    const int lane = threadIdx.x & 31;
    const int wave = threadIdx.x >> 5;
    const int tile = blockIdx.x * WAVES_PER_BLOCK + wave;
    const int tileI = tile / TILES_PER_ROW;
    const int tileJ = tile % TILES_PER_ROW;

    const int sub = lane & 15;          // 0..15
    const int hi  = lane >> 4;          // 0 (K=0,1) or 1 (K=2,3-pad)

    const int row = tileI * 16 + sub;   // A-matrix row handled by this lane
    const int col = tileJ * 16 + sub;   // B-matrix col handled by this lane

    // pos rows (pos is 24 KB total -> L2/L0 resident)
    const float rx = pos[row * 3 + 0];
    const float ry = pos[row * 3 + 1];
    const float rz = pos[row * 3 + 2];
    const float cx = pos[col * 3 + 0];
    const float cy = pos[col * 3 + 1];
    const float cz = pos[col * 3 + 2];

    const float nI = rx * rx + ry * ry + rz * rz;   // |p_row|^2 (row 'sub' of tile)
    const float nJ = cx * cx + cy * cy + cz * cz;   // |p_col|^2 (this lane's column)
    const int   bI = batch[row];
    const int   bJ = batch[col];

    // ISA 16x4 f32 A layout: VGPR0 = K=0 (lanes 0-15) / K=2 (lanes 16-31),
    //                        VGPR1 = K=1 / K=3. K=3 padded with 0.
    v2f a, b;
    a.x = hi ? rz : rx;
    a.y = hi ? 0.0f : ry;
    b.x = hi ? cz : cx;
    b.y = hi ? 0.0f : cy;

    v8f acc = {};
    // D = A(16x4) x B(4x16) + 0  ->  16x16 tile of dot products
    acc = __builtin_amdgcn_wmma_f32_16x16x4_f32(
        /*neg_a=*/false, a, /*neg_b=*/false, b,
        /*c_mod=*/(short)0, acc, /*reuse_a=*/false, /*reuse_b=*/false);

    // fold 1/(2w^2) and log2(e) into one scale -> single v_exp_f32 per RBF;
    // v_rcp_f32 on a wave-uniform value instead of the IEEE divide chain
    const float w = width[0];
    const float kexp = __builtin_amdgcn_rcpf(2.0f * w * w)
                       * 1.44269504088896340736f;

    float cen[NRBF];
#pragma unroll
    for (int r = 0; r < NRBF; ++r) cen[r] = centers[r];

#pragma unroll
    for (int v = 0; v < 8; ++v) {
        const int m = v + (hi << 3);            // row-in-tile of acc[v] (C/D layout)
        const int i = tileI * 16 + m;
        const int j = col;

        const float ni = __shfl(nI, m, 32);     // norm of tile row m (wave32 bcast)
        const int   bi = __shfl(bI, m, 32);

        float d2 = ni + nJ - 2.0f * acc[v];
        d2 = fmaxf(d2, 0.0f);
        const float dist = __builtin_amdgcn_sqrtf(d2);   // raw v_sqrt_f32

        const float maskf =
            ((i != j) && (bi == bJ) && (dist < 8.0f)) ? 1.0f : 0.0f;

        const size_t pidx = (size_t)i * NN + j;
        __builtin_nontemporal_store(maskf, outMask + pidx);

        // 16 contiguous rbf values -> 4 x global_store_b128 (non-temporal)
        v4f ov[4];
        float* ovf = (float*)ov;
#pragma unroll
        for (int r = 0; r < NRBF; ++r) {
            const float t = dist - cen[r];
            ovf[r] = exp2f(-(t * t) * kexp) * maskf;
        }
        v4f* dst = (v4f*)(outRbf + pidx * NRBF);
#pragma unroll
        for (int q = 0; q < 4; ++q)
            __builtin_nontemporal_store(ov[q], dst + q);
    }
}

extern "C" void kernel_launch(void* const* d_in, const int* in_sizes, int n_in,
                              void* d_out, int out_size, void* d_ws, size_t ws_size,
                              hipStream_t stream) {
    const float* pos     = (const float*)d_in[0];  // [N,3] f32
    const int*   batch   = (const int*)d_in[1];    // [N] int
    const float* centers = (const float*)d_in[2];  // [16] f32
    const float* width   = (const float*)d_in[3];  // [1] f32

    float* outMask = (float*)d_out;
    float* outRbf  = (float*)d_out + (size_t)NN * NN;

    const int totalTiles = TILES_PER_ROW * TILES_PER_ROW;      // 16384
    const int blocks = totalTiles / WAVES_PER_BLOCK;           // 2048

    edge_rbf_wmma_kernel<<<blocks, 256, 0, stream>>>(
        pos, batch, centers, width, outMask, outRbf);
}